// AxialClassifier_90855738179587
// MI455X (gfx1250) — compile-verified
//
#include <hip/hip_runtime.h>
#include <math.h>

// Problem constants (from reference)
#define B_   128
#define S_   48
#define D_   8
#define H_   2
#define E_   4
#define HD_  8
#define L_   8
#define NCLS 7
#define HTOT (B_ * S_ * S_ * D_)   // 2,359,296 floats

typedef __attribute__((ext_vector_type(2))) float v2f;
typedef __attribute__((ext_vector_type(8))) float v8f;

// ---------------------------------------------------------------------------
// Wave-level fp32 GEMM built on V_WMMA_F32_16X16X4_F32.
//   C[M,N] = alpha * A[M,K] * B[K,N]      (optionally B given as [N,K], transB)
// Requirements: M, N multiples of 16; K multiple of 4. All operands row-major.
// No guards anywhere -> no exec divergence around the WMMAs.
//
// Fragment layouts per CDNA5 ISA 7.12.2 (wave32):
//   A 16x4 : lane = M + 16*(K>=2), component r -> K = 2*(lane>>4) + r
//   B 4x16 : lane low 4 bits = N, component r -> K = 2*(lane>>4) + r
//   C 16x16: component r -> M = r + 8*(lane>>4), N = lane & 15
// ---------------------------------------------------------------------------
__device__ __forceinline__ void wave_gemm_f32(
    const float* A, int lda,
    const float* B, int ldb, bool transB,
    float* C, int ldc,
    int M, int N, int K, float alpha)
{
  const int lane = threadIdx.x & 31;
  const int half = lane >> 4;   // 0: lanes 0-15, 1: lanes 16-31
  const int l16  = lane & 15;

  for (int m0 = 0; m0 < M; m0 += 16) {
    for (int n0 = 0; n0 < N; n0 += 16) {
      v8f acc = {};
      for (int k0 = 0; k0 < K; k0 += 4) {
        // A fragment (16x4 fp32, 2 VGPRs; consecutive K -> b64 LDS load)
        const int am = m0 + l16;
        const int ak = k0 + (half << 1);
        v2f a;
        a.x = A[am * lda + ak];
        a.y = A[am * lda + ak + 1];
        // B fragment (4x16 fp32, 2 VGPRs)
        const int bn = n0 + l16;
        const int bk = k0 + (half << 1);
        v2f b;
        if (!transB) { b.x = B[bk * ldb + bn]; b.y = B[(bk + 1) * ldb + bn]; }
        else         { b.x = B[bn * ldb + bk]; b.y = B[bn * ldb + bk + 1];   }
        // D = A*B + C  (fp32 WMMA, K=4)
        acc = __builtin_amdgcn_wmma_f32_16x16x4_f32(
            /*neg_a=*/false, a, /*neg_b=*/false, b,
            /*c_mod=*/(short)0, acc, /*reuse_a=*/false, /*reuse_b=*/false);
      }
      // Store C fragment (unguarded; compiler merges into ds_store_2addr)
      #pragma unroll
      for (int r = 0; r < 8; ++r) {
        const int cm = m0 + r + (half << 3);
        const int cn = n0 + l16;
        C[cm * ldc + cn] = acc[r] * alpha;
      }
    }
  }
}

// ---------------------------------------------------------------------------
// Kernel 1: embedding  h = relu(x * enc_w + enc_b) + pos_row + pos_col
// ---------------------------------------------------------------------------
__global__ __launch_bounds__(256) void embed_kernel(
    const float* __restrict__ x, const float* __restrict__ enc_w,
    const float* __restrict__ enc_b, const float* __restrict__ pos_row,
    const float* __restrict__ pos_col, float* __restrict__ h)
{
  int idx = blockIdx.x * blockDim.x + threadIdx.x;
  if (idx >= HTOT) return;
  const int d  = idx & 7;
  const int p  = idx >> 3;            // b*2304 + i*48 + j
  const int j  = p % S_;
  const int i  = (p / S_) % S_;
  const int b  = p / (S_ * S_);
  float v = x[b * (S_ * S_) + i * S_ + j] * enc_w[d] + enc_b[d];
  v = fmaxf(v, 0.0f);
  h[idx] = v + pos_row[i * D_ + d] + pos_col[j * D_ + d];
}

// ---------------------------------------------------------------------------
// Kernel 2: one axial self-attention slice per wave32 block.
// blockIdx -> (b, axis, s).  axis==0: attend over rows (slice = column s),
// axis==1: attend over cols (slice = row s).  Writes o_row / o_col buffers.
// All LDS matrices zero-padded to N=16 so every WMMA tile is full-width.
// ---------------------------------------------------------------------------
__global__ __launch_bounds__(32) void axial_attn_kernel(
    const float* __restrict__ h,
    const float* __restrict__ Wq, const float* __restrict__ Wk,
    const float* __restrict__ Wv, const float* __restrict__ Wo,
    const float* __restrict__ bo,
    float* __restrict__ orow, float* __restrict__ ocol, int layer)
{
  __shared__ float xs[S_ * D_];        // input slice (48 x 8), lda 8
  __shared__ float wqT[8  * 16];       // [d][hd], cols 8..15 zero
  __shared__ float wkT[8  * 16];
  __shared__ float wvT[8  * 16];
  __shared__ float woTp[16 * 16];      // [hd][d], rows/cols 8..15 zero
  __shared__ float qm[S_ * 16];        // (48 x 16), cols 8..15 zero
  __shared__ float km[S_ * 16];
  __shared__ float vm[S_ * 16];
  __shared__ float dots2[S_ * 96];     // [row][head*48 + col]: both heads
  __shared__ float bdiag[96 * 16];     // block-diag [v_h0 ; v_h1], zero-padded
  __shared__ float om[S_ * 16];        // attention output (48 x 16)
  __shared__ float outm[S_ * 16];      // after output projection

  const int lane = threadIdx.x;
  const int bid  = blockIdx.x;
  const int b    = bid / (2 * S_);
  const int rem  = bid % (2 * S_);
  const int axis = rem / S_;
  const int s    = rem % S_;
  const int wbase = (layer * 2 + axis) * (HD_ * D_);   // 8x8 weight block

  // ---- Phase 1: zero padded buffers, load weights + input slice ----
  for (int idx = lane; idx < 8 * 16; idx += 32) {
    wqT[idx] = 0.0f; wkT[idx] = 0.0f; wvT[idx] = 0.0f;
  }
  for (int idx = lane; idx < 16 * 16; idx += 32) woTp[idx] = 0.0f;
  for (int idx = lane; idx < 96 * 16; idx += 32) bdiag[idx] = 0.0f;
  // DS ops from one wave are in-order: fills below safely overwrite zeros.
  for (int idx = lane; idx < 64; idx += 32) {
    const int hd = idx >> 3, d = idx & 7;
    wqT[d * 16 + hd]  = Wq[wbase + idx];           // wqT[d][hd]  = Wq[hd][d]
    wkT[d * 16 + hd]  = Wk[wbase + idx];
    wvT[d * 16 + hd]  = Wv[wbase + idx];
    woTp[hd * 16 + d] = Wo[wbase + d * 8 + hd];    // woTp[hd][d] = Wo[d][hd]
  }
  for (int idx = lane; idx < S_ * D_; idx += 32) {
    const int t = idx >> 3, d = idx & 7;
    const int gi = (axis == 1) ? ((b * S_ + s) * S_ + t)
                               : ((b * S_ + t) * S_ + s);
    xs[idx] = h[gi * D_ + d];
  }
  __syncthreads();

  // ---- Phase 2: Q/K/V projections (48x16) = (48x8) @ (8x16) ----
  wave_gemm_f32(xs, 8, wqT, 16, false, qm, 16, S_, 16, 8, 1.0f);
  wave_gemm_f32(xs, 8, wkT, 16, false, km, 16, S_, 16, 8, 1.0f);
  wave_gemm_f32(xs, 8, wvT, 16, false, vm, 16, S_, 16, 8, 1.0f);
  __syncthreads();

  // ---- Phase 3: dots per head + build block-diagonal V ----
  // dots_h = (q_h @ k_h^T) * E^-0.5 ; (48x48) = (48x4) @ (4x48)
  wave_gemm_f32(qm + 0,  16, km + 0,  16, /*transB=*/true, dots2 + 0,  96, S_, S_, E_, 0.5f);
  wave_gemm_f32(qm + E_, 16, km + E_, 16, /*transB=*/true, dots2 + S_, 96, S_, S_, E_, 0.5f);
  // bdiag rows 0..47 cols 0..3 = v_h0 ; rows 48..95 cols 4..7 = v_h1
  for (int idx = lane; idx < S_ * 8; idx += 32) {
    const int j = idx >> 3, c = idx & 7;
    const int head = c >> 2;
    bdiag[(head * S_ + j) * 16 + c] = vm[j * 16 + c];
  }
  __syncthreads();

  // ---- Phase 4: row-wise softmax (stable), 96 segments of 48 ----
  for (int idx = lane; idx < 96; idx += 32) {
    const int r = idx >> 1, hh = idx & 1;
    float* row = &dots2[r * 96 + hh * S_];
    float mx = row[0];
    for (int j = 1; j < S_; ++j) mx = fmaxf(mx, row[j]);
    float sum = 0.0f;
    for (int j = 0; j < S_; ++j) { float e = __expf(row[j] - mx); row[j] = e; sum += e; }
    const float inv = 1.0f / sum;
    for (int j = 0; j < S_; ++j) row[j] *= inv;
  }
  __syncthreads();

  // ---- Phase 5: o = [attn_h0 | attn_h1] @ blockdiag(v_h0, v_h1) ----
  // (48x16) = (48x96) @ (96x16); cols 0..3 = head0, 4..7 = head1, 8..15 = 0
  wave_gemm_f32(dots2, 96, bdiag, 16, false, om, 16, S_, 16, 96, 1.0f);
  __syncthreads();

  // ---- Phase 6: output projection (48x16) = (48x16) @ (16x16) ----
  wave_gemm_f32(om, 16, woTp, 16, false, outm, 16, S_, 16, 16, 1.0f);
  __syncthreads();

  // ---- Phase 7: scatter (+bias) into the axis-specific output buffer ----
  const float* bop = bo + (layer * 2 + axis) * D_;
  float* obuf = (axis == 1) ? ocol : orow;
  for (int idx = lane; idx < S_ * D_; idx += 32) {
    const int t = idx >> 3, d = idx & 7;
    const int gi = (axis == 1) ? ((b * S_ + s) * S_ + t)
                               : ((b * S_ + t) * S_ + s);
    obuf[gi * D_ + d] = outm[t * 16 + d] + bop[d];
  }
}

// ---------------------------------------------------------------------------
// Kernel 3: h = relu(o_row + o_col)
// ---------------------------------------------------------------------------
__global__ __launch_bounds__(256) void combine_kernel(
    float* __restrict__ h, const float* __restrict__ orow,
    const float* __restrict__ ocol)
{
  int idx = blockIdx.x * blockDim.x + threadIdx.x;
  if (idx >= HTOT) return;
  h[idx] = fmaxf(orow[idx] + ocol[idx], 0.0f);
}

// ---------------------------------------------------------------------------
// Kernel 4: channel-max -> classifier -> softmax. One block per batch.
// ---------------------------------------------------------------------------
__global__ __launch_bounds__(256) void final_kernel(
    const float* __restrict__ h, const float* __restrict__ cls_w,
    const float* __restrict__ cls_b, float* __restrict__ outp)
{
  __shared__ float red[NCLS][256];
  const int b = blockIdx.x, tid = threadIdx.x;
  float part[NCLS];
  #pragma unroll
  for (int c = 0; c < NCLS; ++c) part[c] = 0.0f;

  for (int p = tid; p < S_ * S_; p += 256) {
    const float* hp = &h[(b * S_ * S_ + p) * D_];
    float m = hp[0];
    #pragma unroll
    for (int d = 1; d < D_; ++d) m = fmaxf(m, hp[d]);
    #pragma unroll
    for (int c = 0; c < NCLS; ++c) part[c] += m * cls_w[c * (S_ * S_) + p];
  }
  #pragma unroll
  for (int c = 0; c < NCLS; ++c) red[c][tid] = part[c];
  __syncthreads();
  for (int st = 128; st > 0; st >>= 1) {
    if (tid < st) {
      #pragma unroll
      for (int c = 0; c < NCLS; ++c) red[c][tid] += red[c][tid + st];
    }
    __syncthreads();
  }
  if (tid == 0) {
    float lg[NCLS];
    float mx = -1e30f, sum = 0.0f;
    #pragma unroll
    for (int c = 0; c < NCLS; ++c) { lg[c] = red[c][0] + cls_b[c]; mx = fmaxf(mx, lg[c]); }
    #pragma unroll
    for (int c = 0; c < NCLS; ++c) { lg[c] = __expf(lg[c] - mx); sum += lg[c]; }
    const float inv = 1.0f / sum;
    #pragma unroll
    for (int c = 0; c < NCLS; ++c) outp[b * NCLS + c] = lg[c] * inv;
  }
}

// ---------------------------------------------------------------------------
extern "C" void kernel_launch(void* const* d_in, const int* in_sizes, int n_in,
                              void* d_out, int out_size, void* d_ws, size_t ws_size,
                              hipStream_t stream)
{
  const float* x       = (const float*)d_in[0];
  const float* enc_w   = (const float*)d_in[1];
  const float* enc_b   = (const float*)d_in[2];
  const float* pos_row = (const float*)d_in[3];
  const float* pos_col = (const float*)d_in[4];
  const float* Wq      = (const float*)d_in[5];
  const float* Wk      = (const float*)d_in[6];
  const float* Wv      = (const float*)d_in[7];
  const float* Wo      = (const float*)d_in[8];
  const float* bo      = (const float*)d_in[9];
  const float* cls_w   = (const float*)d_in[10];
  const float* cls_b   = (const float*)d_in[11];

  float* h    = (float*)d_ws;          // (B,48,48,8)
  float* orow = h + HTOT;
  float* ocol = orow + HTOT;

  embed_kernel<<<(HTOT + 255) / 256, 256, 0, stream>>>(
      x, enc_w, enc_b, pos_row, pos_col, h);

  for (int l = 0; l < L_; ++l) {
    axial_attn_kernel<<<B_ * 2 * S_, 32, 0, stream>>>(
        h, Wq, Wk, Wv, Wo, bo, orow, ocol, l);
    combine_kernel<<<(HTOT + 255) / 256, 256, 0, stream>>>(h, orow, ocol);
  }

  final_kernel<<<B_, 256, 0, stream>>>(h, cls_w, cls_b, (float*)d_out);
}